// SESConv2dF_39195871544088
// MI455X (gfx1250) — compile-verified
//
#include <hip/hip_runtime.h>
#include <hip/hip_bf16.h>
#include <math.h>

typedef __attribute__((ext_vector_type(2))) float v2f;
typedef __attribute__((ext_vector_type(8))) float v8f;

#define N128   128
#define NH     65
#define NBINS  (N128 * NH)          // 8320
#define CCH    32
#define BB     32
#define PLANE  (NBINS * CCH * BB)   // 8,519,680 floats per real/imag plane
#define TWO_PI_OVER_N 0.049087385212340517f  // 2*pi/128

// Build a 128-entry twiddle table in LDS: twc[t]=cos(2*pi*t/128), tws[t]=sin(2*pi*t/128)
#define BUILD_TWIDDLES(twc, tws)                                            \
    do {                                                                    \
        if (threadIdx.x < 128) {                                            \
            float s_, c_;                                                   \
            __sincosf(TWO_PI_OVER_N * (float)threadIdx.x, &s_, &c_);        \
            twc[threadIdx.x] = c_;                                          \
            tws[threadIdx.x] = s_;                                          \
        }                                                                   \
        __syncthreads();                                                    \
    } while (0)

// ---------------------------------------------------------------------------
// K0: zero the norm accumulator
// ---------------------------------------------------------------------------
__global__ void k_zero(float* p) {
    if (threadIdx.x == 0 && blockIdx.x == 0) p[0] = 0.0f;
}

// ---------------------------------------------------------------------------
// K1: row-wise real->complex DFT of x.  T[(b*32+c)*128 + row][v], v=0..64
//     e^{-i*2pi*v*j/128} via LDS twiddle table, phase index (v*j) mod 128.
// ---------------------------------------------------------------------------
__global__ void __launch_bounds__(256) k_row_rfft(const float* __restrict__ x,
                                                  float* __restrict__ Tr,
                                                  float* __restrict__ Ti) {
    __shared__ float twc[128], tws[128];
    BUILD_TWIDDLES(twc, tws);
    int idx = blockIdx.x * 256 + threadIdx.x;     // over 32*32*128*65
    if (idx >= PLANE) return;
    int v   = idx % NH;
    int r2  = idx / NH;
    int row = r2 % N128;
    int bc  = r2 / N128;
    const float* xp = x + (bc * N128 + row) * N128;
    float sr = 0.0f, si = 0.0f;
    int ph = 0;
    for (int j = 0; j < N128; ++j) {
        float c = twc[ph], s = tws[ph];
        float xv = xp[j];
        sr += xv * c;
        si -= xv * s;          // e^{-i theta}
        ph = (ph + v) & 127;
    }
    Tr[idx] = sr;
    Ti[idx] = si;
}

// ---------------------------------------------------------------------------
// K2: column-wise complex DFT + transpose into bin layout.
//     X[((u*65+v)*32 + c)*32 + b] = sum_r T[b,c,r,v] * e^{-2pi i u r / 128}
// ---------------------------------------------------------------------------
__global__ void __launch_bounds__(256) k_col_fft(const float* __restrict__ Tr,
                                                 const float* __restrict__ Ti,
                                                 float* __restrict__ Xr,
                                                 float* __restrict__ Xi) {
    __shared__ float twc[128], tws[128];
    BUILD_TWIDDLES(twc, tws);
    int idx = blockIdx.x * 256 + threadIdx.x;     // over PLANE
    if (idx >= PLANE) return;
    int b   = idx & 31;
    int c   = (idx >> 5) & 31;
    int bin = idx >> 10;
    int v   = bin % NH;
    int u   = bin / NH;
    int tbase = ((b * CCH + c) * N128) * NH + v;
    float sr = 0.0f, si = 0.0f;
    int ph = 0;
    for (int r = 0; r < N128; ++r) {
        float c0 = twc[ph], s0 = tws[ph];         // (tr + i*ti) * (c0 - i*s0)
        float tr = Tr[tbase + r * NH];
        float ti = Ti[tbase + r * NH];
        sr += tr * c0 + ti * s0;
        si += ti * c0 - tr * s0;
        ph = (ph + u) & 127;
    }
    Xr[idx] = sr;
    Xi[idx] = si;
}

// ---------------------------------------------------------------------------
// K3: wfft per bin + Frobenius-norm reduction.
//   wfft[u,v][co][ci] = e^{-2pi i (u+v)/128} * sum_{i,j<3} w[co,ci,i,j] e^{+2pi i (u i + v j)/128}
// ---------------------------------------------------------------------------
__global__ void __launch_bounds__(256) k_wfft(const float* __restrict__ w,
                                              float* __restrict__ Wr,
                                              float* __restrict__ Wi,
                                              float* __restrict__ norm) {
    __shared__ float twc[128], tws[128];
    BUILD_TWIDDLES(twc, tws);
    int idx = blockIdx.x * 256 + threadIdx.x;     // over NBINS*32*32
    float sq = 0.0f;
    if (idx < NBINS * CCH * CCH) {
        int ci  = idx & 31;
        int co  = (idx >> 5) & 31;
        int bin = idx >> 10;
        int v   = bin % NH;
        int u   = bin / NH;
        float sr = 0.0f, si = 0.0f;
        for (int i = 0; i < 3; ++i) {
            for (int j = 0; j < 3; ++j) {
                int ph = (u * i + v * j) & 127;   // e^{+i theta}
                float wv = w[((co * CCH + ci) * 3 + i) * 3 + j];
                sr += wv * twc[ph];
                si += wv * tws[ph];
            }
        }
        int ph2 = (u + v) & 127;                  // multiply by e^{-i phi}
        float c2 = twc[ph2], s2 = tws[ph2];
        float rr = sr * c2 + si * s2;
        float ri = si * c2 - sr * s2;
        Wr[idx] = rr;
        Wi[idx] = ri;
        sq = rr * rr + ri * ri;
    }
    __shared__ float red[256];
    red[threadIdx.x] = sq;
    __syncthreads();
    for (int s = 128; s > 0; s >>= 1) {
        if (threadIdx.x < s) red[threadIdx.x] += red[threadIdx.x + s];
        __syncthreads();
    }
    if (threadIdx.x == 0) atomicAdd(norm, red[0]);
}

// ---------------------------------------------------------------------------
// K4: Cayley transform per bin: Q = (I+A)^-1 (I-A), A = g*(W - W^H), g = alpha/||W||_F
//     One wave32 per bin; Gauss-Jordan in LDS (I+A has unit real diagonal -> no pivoting).
//     Writes Q over W in-place.
// ---------------------------------------------------------------------------
__global__ void __launch_bounds__(32) k_cayley(float* __restrict__ Wr,
                                               float* __restrict__ Wi,
                                               const float* __restrict__ normp,
                                               const float* __restrict__ alphap) {
    __shared__ float wr[32][33], wi[32][33];
    __shared__ float Mr[32][33], Mi[32][33];
    __shared__ float Rr[32][33], Ri[32][33];
    int bin = blockIdx.x;
    int t   = threadIdx.x;
    int base = bin * CCH * CCH;
    float g = alphap[0] * rsqrtf(normp[0]);

    for (int j = 0; j < 32; ++j) {
        wr[t][j] = Wr[base + t * 32 + j];
        wi[t][j] = Wi[base + t * 32 + j];
    }
    __syncthreads();
    for (int j = 0; j < 32; ++j) {
        float ar = g * (wr[t][j] - wr[j][t]);   // Re(W_tj - conj(W_jt))
        float ai = g * (wi[t][j] + wi[j][t]);   // Im(W_tj - conj(W_jt))
        float d  = (t == j) ? 1.0f : 0.0f;
        Mr[t][j] = d + ar;  Mi[t][j] = ai;      // I + A
        Rr[t][j] = d - ar;  Ri[t][j] = -ai;     // I - A
    }
    __syncthreads();

    for (int k = 0; k < 32; ++k) {
        float pr = Mr[k][k], pi = Mi[k][k];
        float den = 1.0f / (pr * pr + pi * pi);
        float ir = pr * den, ii = -pi * den;    // 1/pivot
        float fr = 0.0f, fi = 0.0f;
        if (t != k) {
            float mr = Mr[t][k], mi = Mi[t][k];
            fr = mr * ir - mi * ii;
            fi = mr * ii + mi * ir;
        }
        __syncthreads();
        if (t != k) {
            for (int j = 0; j < 32; ++j) {
                float mkr = Mr[k][j], mki = Mi[k][j];
                Mr[t][j] -= fr * mkr - fi * mki;
                Mi[t][j] -= fr * mki + fi * mkr;
                float rkr = Rr[k][j], rki = Ri[k][j];
                Rr[t][j] -= fr * rkr - fi * rki;
                Ri[t][j] -= fr * rki + fi * rkr;
            }
        }
        __syncthreads();
    }
    float pr = Mr[t][t], pi = Mi[t][t];
    float den = 1.0f / (pr * pr + pi * pi);
    float ir = pr * den, ii = -pi * den;
    for (int j = 0; j < 32; ++j) {
        float qr = Rr[t][j] * ir - Ri[t][j] * ii;
        float qi = Rr[t][j] * ii + Ri[t][j] * ir;
        Wr[base + t * 32 + j] = qr;
        Wi[base + t * 32 + j] = qi;
    }
}

// ---------------------------------------------------------------------------
// K5: per-bin complex GEMM Y = Q @ X via V_WMMA_F32_16X16X4_F32 (fp32, full accuracy).
//     4 waves/block, one 16x16 output tile per wave, K=32 in 8 wmma k-steps.
//     Yr = Qr*Xr - Qi*Xi ; Yi = Qr*Xi + Qi*Xr  (Qi negated in registers for the Yr term,
//     since f32 WMMA NEG bits only cover the C operand).  Writes Y over X in-place.
// ---------------------------------------------------------------------------
__global__ void __launch_bounds__(128) k_qx_wmma(const float* __restrict__ Qr,
                                                 const float* __restrict__ Qi,
                                                 float* __restrict__ Xr,
                                                 float* __restrict__ Xi) {
    const int bin  = blockIdx.x;
    const int wave = threadIdx.x >> 5;
    const int lane = threadIdx.x & 31;
    const int m0 = (wave & 1) * 16;
    const int n0 = (wave >> 1) * 16;
    const int base = bin * CCH * BB;
    const int M  = m0 + (lane & 15);   // A-matrix row owned by this lane
    const int Nn = n0 + (lane & 15);   // B/C column owned by this lane
    const int hi = (lane >> 4) & 1;    // half-wave select (K split / M+8 split)

    v8f accR = {};
    v8f accI = {};
    for (int k0 = 0; k0 < 32; k0 += 4) {
        const int ka = k0 + hi * 2;    // A: lanes 0-15 hold K,K+1; lanes 16-31 hold K+2,K+3
        v2f ar, ai, ain, br, bi;
        ar.x = Qr[base + M * 32 + ka];       ar.y = Qr[base + M * 32 + ka + 1];
        ai.x = Qi[base + M * 32 + ka];       ai.y = Qi[base + M * 32 + ka + 1];
        ain.x = -ai.x;                       ain.y = -ai.y;
        br.x = Xr[base + ka * 32 + Nn];      br.y = Xr[base + (ka + 1) * 32 + Nn];
        bi.x = Xi[base + ka * 32 + Nn];      bi.y = Xi[base + (ka + 1) * 32 + Nn];
        accR = __builtin_amdgcn_wmma_f32_16x16x4_f32(false, ar,  false, br, (short)0, accR, false, false);
        accR = __builtin_amdgcn_wmma_f32_16x16x4_f32(false, ain, false, bi, (short)0, accR, false, false);
        accI = __builtin_amdgcn_wmma_f32_16x16x4_f32(false, ar,  false, bi, (short)0, accI, false, false);
        accI = __builtin_amdgcn_wmma_f32_16x16x4_f32(false, ai,  false, br, (short)0, accI, false, false);
    }
    __syncthreads();  // all waves finished reading X for this bin before overwrite
    const int mrow = m0 + hi * 8;      // C/D layout: VGPR r -> M=r (lanes 0-15) / M=r+8 (16-31)
#pragma unroll
    for (int e = 0; e < 8; ++e) {
        int off = base + (mrow + e) * 32 + Nn;
        Xr[off] = accR[e];
        Xi[off] = accI[e];
    }
}

// ---------------------------------------------------------------------------
// K6: inverse DFT along u (complex, with 1/128), back to [b,c,u',v] layout.
// ---------------------------------------------------------------------------
__global__ void __launch_bounds__(256) k_icol_fft(const float* __restrict__ Yr,
                                                  const float* __restrict__ Yi,
                                                  float* __restrict__ Zr,
                                                  float* __restrict__ Zi) {
    __shared__ float twc[128], tws[128];
    BUILD_TWIDDLES(twc, tws);
    int idx = blockIdx.x * 256 + threadIdx.x;     // over PLANE: ((b*32+c)*128+u')*65+v
    if (idx >= PLANE) return;
    int v  = idx % NH;
    int r2 = idx / NH;
    int up = r2 % N128;
    int bc = r2 / N128;
    int b  = bc / 32;
    int c  = bc % 32;
    float sr = 0.0f, si = 0.0f;
    int ph = 0;
    for (int u = 0; u < N128; ++u) {
        int yidx = ((u * NH + v) * CCH + c) * BB + b;
        float c0 = twc[ph], s0 = tws[ph];         // (yr + i*yi) * (c0 + i*s0)
        float yr = Yr[yidx], yi = Yi[yidx];
        sr += yr * c0 - yi * s0;
        si += yr * s0 + yi * c0;
        ph = (ph + up) & 127;
    }
    Zr[idx] = sr * (1.0f / 128.0f);
    Zi[idx] = si * (1.0f / 128.0f);
}

// ---------------------------------------------------------------------------
// K7: row-wise inverse real DFT (irfft) + bias.
// ---------------------------------------------------------------------------
__global__ void __launch_bounds__(256) k_irow_irfft(const float* __restrict__ Zr,
                                                    const float* __restrict__ Zi,
                                                    const float* __restrict__ bias,
                                                    float* __restrict__ out) {
    __shared__ float twc[128], tws[128];
    BUILD_TWIDDLES(twc, tws);
    int idx = blockIdx.x * 256 + threadIdx.x;     // over 32*32*128*128
    if (idx >= BB * CCH * N128 * N128) return;
    int j  = idx & 127;
    int r2 = idx >> 7;          // bc*128 + row
    int bc = r2 >> 7;
    int c  = bc & 31;
    int zbase = r2 * NH;
    float acc = Zr[zbase];                                   // v = 0
    acc += ((j & 1) ? -1.0f : 1.0f) * Zr[zbase + 64];        // v = 64 (Nyquist)
    int ph = j & 127;
    for (int v = 1; v < 64; ++v) {
        acc += 2.0f * (Zr[zbase + v] * twc[ph] - Zi[zbase + v] * tws[ph]);
        ph = (ph + j) & 127;
    }
    out[idx] = acc * (1.0f / 128.0f) + bias[c];
}

// ---------------------------------------------------------------------------
// launch
// ---------------------------------------------------------------------------
extern "C" void kernel_launch(void* const* d_in, const int* in_sizes, int n_in,
                              void* d_out, int out_size, void* d_ws, size_t ws_size,
                              hipStream_t stream) {
    const float* x      = (const float*)d_in[0];   // [32,32,128,128]
    const float* weight = (const float*)d_in[1];   // [32,32,3,3]
    const float* bias   = (const float*)d_in[2];   // [32]
    const float* alpha  = (const float*)d_in[3];   // [1]
    float* out = (float*)d_out;

    float* ws = (float*)d_ws;
    // BUF0: X (later overwritten in-place by Y = Q@X)
    float* Xr = ws;
    float* Xi = ws + PLANE;
    // BUF1: T (row-FFT temp) -> W (wfft) -> Q (Cayley) -> Z (inverse column FFT)
    float* Br = ws + 2 * PLANE;
    float* Bi = ws + 3 * PLANE;
    float* normp = ws + 4 * PLANE;   // 1 float

    const int g1 = (PLANE + 255) / 256;                       // 33280
    const int g2 = (BB * CCH * N128 * N128 + 255) / 256;      // 65536

    k_zero<<<1, 32, 0, stream>>>(normp);
    k_row_rfft<<<g1, 256, 0, stream>>>(x, Br, Bi);            // T in BUF1
    k_col_fft<<<g1, 256, 0, stream>>>(Br, Bi, Xr, Xi);        // X in BUF0
    k_wfft<<<g1, 256, 0, stream>>>(weight, Br, Bi, normp);    // W over T in BUF1
    k_cayley<<<NBINS, 32, 0, stream>>>(Br, Bi, normp, alpha); // Q over W in BUF1
    k_qx_wmma<<<NBINS, 128, 0, stream>>>(Br, Bi, Xr, Xi);     // Y over X in BUF0 (WMMA)
    k_icol_fft<<<g1, 256, 0, stream>>>(Xr, Xi, Br, Bi);       // Z over Q in BUF1
    k_irow_irfft<<<g2, 256, 0, stream>>>(Br, Bi, bias, out);
    (void)in_sizes; (void)n_in; (void)out_size; (void)ws_size;
}